// BaseViewTransform_26310969655942
// MI455X (gfx1250) — compile-verified
//
#include <hip/hip_runtime.h>
#include <hip/hip_bf16.h>
#include <stdint.h>

// ---- problem constants (from reference) ----
#define NCAM 6
#define DDIM 118
#define HDIM 32
#define WDIM 88
#define CDIM 80
#define XDIM 360
#define YDIM 360
#define OUT_XY (XDIM * YDIM)                 // 129600

// ---- tiling: one tile = 32 h-rows x 4 w-columns x 80 ch ----
#define THREADS 160                          // 5 waves
#define WTILE 4
#define PT_BYTES (CDIM * 4)                  // 320
#define ROW_BYTES (WTILE * PT_BYTES)         // 1280
#define ROW_ELEMS (WTILE * CDIM)             // 320 (4B elements per tile row)
#define TILE_BYTES (HDIM * ROW_BYTES)        // 40960
#define PLANE_ROW_BYTES (WDIM * PT_BYTES)    // 28160 (one h row of a (n,d) plane)
#define PLANE_ROW_ELEMS (WDIM * CDIM)        // 7040
#define WBLOCKS (WDIM / WTILE)               // 22
#define NUM_TILES (NCAM * DDIM * WBLOCKS)    // 15576

// ---------------- CDNA5 TDM helpers ----------------
typedef uint32_t u32x4 __attribute__((ext_vector_type(4)));
typedef int      i32x4 __attribute__((ext_vector_type(4)));
typedef int      i32x8 __attribute__((ext_vector_type(8)));

template <int N>
__device__ __forceinline__ void wait_tensorcnt() {
#if defined(__has_builtin) && __has_builtin(__builtin_amdgcn_s_wait_tensorcnt)
  __builtin_amdgcn_s_wait_tensorcnt(N);
#else
  asm volatile("s_wait_tensorcnt %0" ::"i"(N) : "memory");
#endif
}

// One TDM DMA: 2D tile, 32 rows x 1280B, row stride 28160B, global -> LDS.
__device__ __forceinline__ void tdm_load_tile(uint64_t gaddr, uint32_t lds_addr) {
  u32x4 g0;
  g0[0] = 1u;                                       // count=1 (valid user D#)
  g0[1] = lds_addr;                                 // LDS byte address
  g0[2] = (uint32_t)gaddr;                          // global_addr[31:0]
  g0[3] = (uint32_t)(gaddr >> 32) | (2u << 30);     // global_addr[56:32] | type=2

  i32x8 g1;
  g1[0] = (int)(2u << 16);                          // workgroup_mask=0 | data_size=2 (4B)
  g1[1] = (int)((uint32_t)ROW_ELEMS << 16);         // tensor_dim0[15:0] (=320)
  g1[2] = (int)((uint32_t)HDIM << 16);              // tensor_dim0 hi=0 | tensor_dim1[15:0]=32
  g1[3] = (int)((uint32_t)ROW_ELEMS << 16);         // tensor_dim1 hi=0 | tile_dim0=320
  g1[4] = (int)HDIM;                                // tile_dim1=32 | tile_dim2=0
  g1[5] = (int)PLANE_ROW_ELEMS;                     // tensor_dim0_stride[31:0]=7040
  g1[6] = 0;                                        // stride0 hi | stride1 lo
  g1[7] = 0;                                        // stride1 hi

  i32x4 gz = {0, 0, 0, 0};
#if __has_include(<hip/amd_detail/amd_gfx1250_TDM.h>)
  i32x8 gz8 = {0, 0, 0, 0, 0, 0, 0, 0};
  __builtin_amdgcn_tensor_load_to_lds(g0, g1, gz, gz, gz8, 0);
#else
  __builtin_amdgcn_tensor_load_to_lds(g0, g1, gz, gz, 0);
#endif
}

// ---------------- small 3x3 math ----------------
__device__ __forceinline__ void inv3(const float* m, float* o) {
  float a = m[0], b = m[1], c = m[2];
  float d = m[3], e = m[4], f = m[5];
  float g = m[6], h = m[7], i = m[8];
  float A = e * i - f * h;
  float B = -(d * i - f * g);
  float C = d * h - e * g;
  float r = 1.0f / (a * A + b * B + c * C);
  o[0] = A * r;  o[1] = -(b * i - c * h) * r;  o[2] = (b * f - c * e) * r;
  o[3] = B * r;  o[4] = (a * i - c * g) * r;   o[5] = -(a * f - c * d) * r;
  o[6] = C * r;  o[7] = -(a * h - b * g) * r;  o[8] = (a * e - b * d) * r;
}

__device__ __forceinline__ void mm3(const float* a, const float* b, float* o) {
#pragma unroll
  for (int i = 0; i < 3; ++i)
#pragma unroll
    for (int j = 0; j < 3; ++j)
      o[i * 3 + j] = a[i * 3] * b[j] + a[i * 3 + 1] * b[3 + j] + a[i * 3 + 2] * b[6 + j];
}

// ---------------- kernels ----------------
__global__ void zero_out_kernel(float* __restrict__ out, int n) {
  int n4 = n >> 2;
  float4 z = make_float4(0.f, 0.f, 0.f, 0.f);
  for (int i = blockIdx.x * blockDim.x + threadIdx.x; i < n4;
       i += gridDim.x * blockDim.x)
    ((float4*)out)[i] = z;
  int tail = n & 3;
  if (blockIdx.x == 0 && threadIdx.x < (unsigned)tail) out[n4 * 4 + threadIdx.x] = 0.f;
}

__global__ void __launch_bounds__(THREADS)
bev_scatter_kernel(const float* __restrict__ x,
                   const float* __restrict__ c2l,
                   const float* __restrict__ intr,
                   const float* __restrict__ aug,
                   const float* __restrict__ laug,
                   float* __restrict__ out) {
  __shared__ __align__(16) unsigned char s_buf[2 * TILE_BYTES];  // 81920 B
  __shared__ int s_vox[WTILE * HDIM];                            // voxel id or -1
  __shared__ float s_cam[NCAM * 24];  // [0..8] invPostRot, [9..17] A, [18..20] bvec, [21..23] postTrans

  const int t = threadIdx.x;
  const bool issuer = (t < 32);  // wave 0 drives the TDM

  // per-block camera parameter setup (tiny)
  if (t < NCAM) {
    const int n = t;
    float R[9], K[9], P[9], E[9], ct[3], pt[3], et[3];
#pragma unroll
    for (int i = 0; i < 3; ++i) {
#pragma unroll
      for (int j = 0; j < 3; ++j) {
        R[i * 3 + j] = c2l[n * 16 + i * 4 + j];
        K[i * 3 + j] = intr[n * 16 + i * 4 + j];
        P[i * 3 + j] = aug[n * 16 + i * 4 + j];
        E[i * 3 + j] = laug[i * 4 + j];
      }
      ct[i] = c2l[n * 16 + i * 4 + 3];
      pt[i] = aug[n * 16 + i * 4 + 3];
      et[i] = laug[i * 4 + 3];
    }
    float Kinv[9], Pinv[9], RK[9], A[9];
    inv3(K, Kinv);
    inv3(P, Pinv);
    mm3(R, Kinv, RK);
    mm3(E, RK, A);
    float* o = &s_cam[n * 24];
#pragma unroll
    for (int i = 0; i < 9; ++i) o[i] = Pinv[i];
#pragma unroll
    for (int i = 0; i < 9; ++i) o[9 + i] = A[i];
#pragma unroll
    for (int i = 0; i < 3; ++i)
      o[18 + i] = E[i * 3] * ct[0] + E[i * 3 + 1] * ct[1] + E[i * 3 + 2] * ct[2] + et[i];
#pragma unroll
    for (int i = 0; i < 3; ++i) o[21 + i] = pt[i];
  }

  int tile = blockIdx.x;
  if (tile >= NUM_TILES) return;  // block-uniform

  const uint32_t lbase = (uint32_t)(uintptr_t)&s_buf[0];

  // tile -> (n, d, wblock); global byte addr of (n, d, h=0, w=w0)
  auto tile_gbase = [&](int tl) -> uint64_t {
    int nd = tl / WBLOCKS;            // n*DDIM + d
    int wb = tl - nd * WBLOCKS;
    return (uint64_t)(uintptr_t)x +
           (uint64_t)nd * (HDIM * (uint64_t)PLANE_ROW_BYTES) +
           (uint64_t)wb * (WTILE * PT_BYTES);
  };

  if (issuer) tdm_load_tile(tile_gbase(tile), lbase);  // prefetch first tile
  int buf = 0;

  for (; tile < NUM_TILES; tile += gridDim.x) {
    const int nxt = tile + gridDim.x;
    const int nd = tile / WBLOCKS;
    const int wb = tile - nd * WBLOCKS;
    const int n = nd / DDIM;
    const int d = nd - n * DDIM;
    const int w0 = wb * WTILE;

    if (issuer && nxt < NUM_TILES)
      tdm_load_tile(tile_gbase(nxt), lbase + (buf ^ 1) * TILE_BYTES);

    // ---- phase 1: one voxel id per thread (128 points), overlaps the DMA ----
    if (t < WTILE * HDIM) {
      const int col = t >> 5;          // 0..3
      const int h = t & 31;
      const int w = w0 + col;
      const float* cp = &s_cam[n * 24];
      float xi = (float)w * (703.0f / 87.0f);
      float yi = (float)h * (255.0f / 31.0f);
      float dep = 1.0f + 0.5f * (float)d;
      float p0x = xi - cp[21], p0y = yi - cp[22], p0z = dep - cp[23];
      float q0 = cp[0] * p0x + cp[1] * p0y + cp[2] * p0z;
      float q1 = cp[3] * p0x + cp[4] * p0y + cp[5] * p0z;
      float q2 = cp[6] * p0x + cp[7] * p0y + cp[8] * p0z;
      float m0 = q0 * q2, m1 = q1 * q2, m2 = q2;
      float px = cp[9]  * m0 + cp[10] * m1 + cp[11] * m2 + cp[18];
      float py = cp[12] * m0 + cp[13] * m1 + cp[14] * m2 + cp[19];
      float pz = cp[15] * m0 + cp[16] * m1 + cp[17] * m2 + cp[20];
      // truncation toward zero matches jnp .astype(int32)
      int cx = (int)((px + 54.0f) / 0.3f);
      int cy = (int)((py + 54.0f) / 0.3f);
      int cz = (int)((pz + 10.0f) / 20.0f);
      bool kept = (cx >= 0) & (cx < XDIM) & (cy >= 0) & (cy < YDIM) & (cz == 0);
      s_vox[t] = kept ? (cx * YDIM + cy) : -1;
    }

    // wave 0: current tile's DMA done (TDM ops in-order per wave), next may fly
    if (issuer) {
      if (nxt < NUM_TILES) wait_tensorcnt<1>();
      else                 wait_tensorcnt<0>();
    }
    __syncthreads();  // releases tile data + s_vox + (first iter) s_cam

    // ---- phase 2: run-merged accumulation ----
    // thread -> (col, q, half): 4 * 20 * 2 = 160
    {
      const int col = t / 40;
      const int sub = t - col * 40;
      const int q = sub % 20;          // channel quad: channels q*4..q*4+3
      const int half = sub / 20;       // h in [half*16, half*16+16)
      const unsigned base =
          (unsigned)buf * TILE_BYTES + (unsigned)col * PT_BYTES + (unsigned)q * 16u;
      float* oq = out + (size_t)(q * 4) * OUT_XY;

      float4 acc = make_float4(0.f, 0.f, 0.f, 0.f);
      int cur = -1;
      for (int h = half * 16; h < half * 16 + 16; ++h) {
        int vox = s_vox[col * HDIM + h];
        const float4 v = *(const float4*)(&s_buf[base + (unsigned)h * ROW_BYTES]);
        if (vox == cur) {
          if (cur >= 0) { acc.x += v.x; acc.y += v.y; acc.z += v.z; acc.w += v.w; }
        } else {
          if (cur >= 0) {
            unsafeAtomicAdd(oq + cur, acc.x);
            unsafeAtomicAdd(oq + OUT_XY + cur, acc.y);
            unsafeAtomicAdd(oq + 2 * OUT_XY + cur, acc.z);
            unsafeAtomicAdd(oq + 3 * OUT_XY + cur, acc.w);
          }
          acc = v;
          cur = vox;
        }
      }
      if (cur >= 0) {
        unsafeAtomicAdd(oq + cur, acc.x);
        unsafeAtomicAdd(oq + OUT_XY + cur, acc.y);
        unsafeAtomicAdd(oq + 2 * OUT_XY + cur, acc.z);
        unsafeAtomicAdd(oq + 3 * OUT_XY + cur, acc.w);
      }
    }

    __syncthreads();  // all reads of this buffer done before it is re-filled
    buf ^= 1;
  }
}

// ---------------- host launcher ----------------
extern "C" void kernel_launch(void* const* d_in, const int* in_sizes, int n_in,
                              void* d_out, int out_size, void* d_ws, size_t ws_size,
                              hipStream_t stream) {
  const float* x     = (const float*)d_in[0];
  const float* c2l   = (const float*)d_in[1];
  const float* intr  = (const float*)d_in[2];
  const float* aug   = (const float*)d_in[3];
  const float* laug  = (const float*)d_in[4];
  float* out = (float*)d_out;

  zero_out_kernel<<<1024, 256, 0, stream>>>(out, out_size);
  bev_scatter_kernel<<<4096, THREADS, 0, stream>>>(x, c2l, intr, aug, laug, out);
}